// MPSClassifier_3504693314226
// MI455X (gfx1250) — compile-verified
//
#include <hip/hip_runtime.h>
#include <math.h>

// Problem constants (from reference): B=2048, N=784 sites, D=10 state, F=2, OUT=10
#define NB      2048
#define NSITES  784
#define NMID    783   // number of mid cores (sites 1..783)
#define EPSN    1e-12f

typedef float v2f __attribute__((ext_vector_type(2)));
typedef float v8f __attribute__((ext_vector_type(8)));

// D = A(16x4 f32) * B(4x16 f32) + C(16x16 f32)   -- CDNA5 V_WMMA_F32_16X16X4_F32
__device__ __forceinline__ v8f wmma4(v2f a, v2f b, v8f c) {
  return __builtin_amdgcn_wmma_f32_16x16x4_f32(
      /*neg_a=*/false, a, /*neg_b=*/false, b,
      /*c_mod=*/(short)0, c, /*reuse_a=*/false, /*reuse_b=*/false);
}

// Re-layout a 16x16 f32 C/D-layout tile (rows 0..9 valid, rest 0) into the three
// 4x16 B-operand chunks covering K rows 0..11 (rows 10,11 are zero padding).
// C/D layout: row r -> vgpr r%8, half r/8.  B chunk kc, vgpr j:
//   lanes 0-15 hold row 4kc+j, lanes 16-31 hold row 4kc+2+j.
__device__ __forceinline__ void relayout(const v8f& t, bool lo,
                                         v2f& u0, v2f& u1, v2f& u2) {
  float s0 = __shfl_xor(t[0], 16, 32);
  float s1 = __shfl_xor(t[1], 16, 32);
  float s2 = __shfl_xor(t[2], 16, 32);
  float s3 = __shfl_xor(t[3], 16, 32);
  float s6 = __shfl_xor(t[6], 16, 32);
  float s7 = __shfl_xor(t[7], 16, 32);
  u0.x = lo ? t[0] : s2;   u0.y = lo ? t[1] : s3;   // rows 0,1 | 2,3
  u1.x = lo ? t[4] : s6;   u1.y = lo ? t[5] : s7;   // rows 4,5 | 6,7
  u2.x = lo ? s0 : 0.0f;   u2.y = lo ? s1 : 0.0f;   // rows 8,9 | 10,11(=0)
}

// Prep: swizzle cores_mid[n][d][f][h] into exact WMMA A-operand lane order.
// For GEMM  Y = coreT(16xK) * VT : A-operand chunk kc, lane, vgpr j holds
//   coreT[M=lane%16][k=4kc+j+2*(lane/16)] = core[d=4kc+j+2*(lane/16)][h=lane%16]
// aop = A-matrix (f=0), bop = B-A difference matrix. Zero padding baked in.
__global__ __launch_bounds__(256) void mps_prep(const float* __restrict__ cm,
                                                v2f* __restrict__ aop,
                                                v2f* __restrict__ bop) {
  int t = blockIdx.x * blockDim.x + threadIdx.x;
  if (t >= NMID * 96) return;           // 96 = 3 chunks * 32 lanes
  int n    = t / 96;
  int r    = t - n * 96;
  int kc   = r >> 5;
  int lane = r & 31;
  int hi   = lane >> 4;
  int h    = lane & 15;
  v2f av = {0.0f, 0.0f}, bv = {0.0f, 0.0f};
#pragma unroll
  for (int j = 0; j < 2; ++j) {
    int d = 4 * kc + j + 2 * hi;
    float a = 0.0f, bd = 0.0f;
    if (d < 10 && h < 10) {
      size_t base = (((size_t)n * 10 + d) * 2) * 10 + h;  // [n][d][0][h]
      a  = cm[base];
      bd = cm[base + 10] - a;                              // [n][d][1][h] - a
    }
    if (j == 0) { av.x = a; bv.x = bd; } else { av.y = a; bv.y = bd; }
  }
  aop[t] = av;
  bop[t] = bv;
}

// Main: one wave per 16 batch elements. VT (state x batch, 16x16 padded) lives
// in WMMA C/D layout. Per site: 6 coalesced b64 operand loads, 1 x load,
// relayout (6 swaps), 6 v_wmma_f32_16x16x4_f32, 8 FMA combine.
__global__ __launch_bounds__(256) void mps_main(const float* __restrict__ x,
                                                const float* __restrict__ core0,
                                                const float* __restrict__ cls,
                                                const v2f* __restrict__ aop,
                                                const v2f* __restrict__ bop,
                                                float* __restrict__ out) {
  const int  lane  = threadIdx.x & 31;
  const int  gwave = (blockIdx.x * blockDim.x + threadIdx.x) >> 5;
  const int  bbase = gwave * 16;
  const bool lo    = lane < 16;
  const int  col   = lane & 15;
  const int  hi    = lane >> 4;

  const float* xcol = x + (size_t)(bbase + col) * NSITES;

  // Init head: VT[h, b] = core0[0][h] + x[b,0]*(core0[1][h]-core0[0][h])
  float x0 = xcol[0];
  v8f t;
#pragma unroll
  for (int k = 0; k < 8; ++k) {
    int h = k + 8 * hi;
    float a = (h < 10) ? core0[h]      : 0.0f;
    float b = (h < 10) ? core0[10 + h] : 0.0f;
    t[k] = a + x0 * (b - a);
  }
  float lognorm = 0.0f;
  const v8f z = {};

  for (int n = 1; n < NSITES; ++n) {
    const int m = n - 1;
    const v2f* ap = aop + (size_t)m * 96;
    const v2f* bp = bop + (size_t)m * 96;
    v2f a0 = ap[lane], a1 = ap[32 + lane], a2 = ap[64 + lane];
    v2f b0 = bp[lane], b1 = bp[32 + lane], b2 = bp[64 + lane];
    float xb = xcol[n];
    if (m + 8 < NMID) {   // global_prefetch_b8 for upcoming operand blocks
      __builtin_prefetch(aop + (size_t)(m + 8) * 96 + lane, 0, 1);
      __builtin_prefetch(bop + (size_t)(m + 8) * 96 + lane, 0, 1);
    }

    v2f u0, u1, u2;
    relayout(t, lo, u0, u1, u2);

    v8f y1 = wmma4(a2, u2, wmma4(a1, u1, wmma4(a0, u0, z)));  // A^T @ VT
    v8f y2 = wmma4(b2, u2, wmma4(b1, u1, wmma4(b0, u0, z)));  // (B-A)^T @ VT

#pragma unroll
    for (int k = 0; k < 8; ++k) t[k] = y1[k] + xb * y2[k];

    if ((n & 15) == 0 || n == NSITES - 1) {   // periodic + final renorm
      float ss = 0.0f;
#pragma unroll
      for (int k = 0; k < 8; ++k) ss += t[k] * t[k];
      ss += __shfl_xor(ss, 16, 32);           // column total (per batch b)
      float nrm = fmaxf(sqrtf(ss), EPSN);
      lognorm += logf(nrm);
      float inv = 1.0f / nrm;
#pragma unroll
      for (int k = 0; k < 8; ++k) t[k] *= inv;
    }
  }

  // Classifier GEMM: L[o,b] = sum_h cls[o][h] * VT[h,b]  (gathered A-operand)
  v2f c0, c1, c2;
  {
    int o = col;
#pragma unroll
    for (int kc = 0; kc < 3; ++kc) {
      int k0 = 4 * kc + 2 * hi;
      float vx = (o < 10 && k0     < 10) ? cls[o * 10 + k0]     : 0.0f;
      float vy = (o < 10 && k0 + 1 < 10) ? cls[o * 10 + k0 + 1] : 0.0f;
      if (kc == 0) { c0.x = vx; c0.y = vy; }
      else if (kc == 1) { c1.x = vx; c1.y = vy; }
      else { c2.x = vx; c2.y = vy; }
    }
  }
  v2f u0, u1, u2;
  relayout(t, lo, u0, u1, u2);
  v8f L = wmma4(c2, u2, wmma4(c1, u1, wmma4(c0, u0, z)));

  // logits[b,o] = L[o,b] + lognorm[b]; rows o = k+8*hi in C/D layout
  int b = bbase + col;
  if (lo) {
#pragma unroll
    for (int k = 0; k < 8; ++k) out[(size_t)b * 10 + k] = L[k] + lognorm;
  } else {
    out[(size_t)b * 10 + 8] = L[0] + lognorm;
    out[(size_t)b * 10 + 9] = L[1] + lognorm;
  }
}

extern "C" void kernel_launch(void* const* d_in, const int* in_sizes, int n_in,
                              void* d_out, int out_size, void* d_ws, size_t ws_size,
                              hipStream_t stream) {
  const float* x     = (const float*)d_in[0];  // (2048, 784)
  const float* core0 = (const float*)d_in[1];  // (1, 2, 10)
  const float* cm    = (const float*)d_in[2];  // (783, 10, 2, 10)
  const float* cls   = (const float*)d_in[3];  // (10, 10)
  float* out = (float*)d_out;                  // (2048, 10)

  // Workspace: pre-swizzled WMMA A-operands, 2 * 783*96 * 8B ~= 1.2 MB
  v2f* aop = (v2f*)d_ws;
  v2f* bop = aop + (size_t)NMID * 96;

  int prepN = NMID * 96;
  mps_prep<<<(prepN + 255) / 256, 256, 0, stream>>>(cm, aop, bop);

  // 2048 batches / 16 per wave = 128 waves = 16 blocks of 256 threads
  mps_main<<<NB / 16 / 8, 256, 0, stream>>>(x, core0, cls, aop, bop, out);
}